// SIGReg_74474732912931
// MI455X (gfx1250) — compile-verified
//
#include <hip/hip_runtime.h>

typedef __attribute__((ext_vector_type(2))) float v2f;
typedef __attribute__((ext_vector_type(8))) float v8f;

// Problem sizes from the reference
constexpr int T = 32;      // time steps (outer)
constexpr int B = 128;     // batch (reduced)
constexpr int D = 512;     // feature dim (GEMM K)
constexpr int M = 1024;    // projections (GEMM N)
constexpr int K = 17;      // trapezoid knots
constexpr int MT = 64;     // M tile per block
constexpr int KC = 64;     // D chunk staged in LDS
constexpr int LZS = KC + 4; // z-tile row stride: 4-bank shift/row -> conflict-free A-frag b64 loads
constexpr int LAS = MT + 8; // A-tile row stride: 2-row offset = 16 banks -> halves disjoint on B loads,
                            // and 288B rows keep ds_store_b128 staging 16B-aligned

__global__ __launch_bounds__(256) void col_inv_norm_kernel(const float* __restrict__ A,
                                                           float* __restrict__ invn) {
    int m = blockIdx.x * blockDim.x + threadIdx.x;
    if (m >= M) return;
    float s = 0.f;
    for (int d = 0; d < D; ++d) {
        float v = A[(size_t)d * M + m];
        s = fmaf(v, v, s);
    }
    invn[m] = 1.0f / fmaxf(sqrtf(s), 1e-12f);
}

__global__ void zero_kernel(float* out) { out[0] = 0.0f; }

__global__ __launch_bounds__(256) void sigreg_fused_kernel(const float* __restrict__ z,
                                                           const float* __restrict__ A,
                                                           const float* __restrict__ invn,
                                                           float* __restrict__ out) {
    __shared__ float lz[B * LZS];   // z chunk (B x KC); reused afterwards as proj tile (B x MT)
    __shared__ float la[KC * LAS];  // normalized A chunk (KC x MT), padded stride
    __shared__ float linv[MT];
    __shared__ float red[256];

    const int tid  = threadIdx.x;
    const int wave = tid >> 5;
    const int lane = tid & 31;
    const int half = lane >> 4;     // which 16-lane half
    const int l16  = lane & 15;
    const int t    = blockIdx.x;            // 0..31
    const int m0   = blockIdx.y * MT;       // M tile base

    if (tid < MT) linv[tid] = invn[m0 + tid];

    const v8f zero8 = {0.f, 0.f, 0.f, 0.f, 0.f, 0.f, 0.f, 0.f};
    v8f c[4] = {zero8, zero8, zero8, zero8};

    const float* zt = z + (size_t)t * B * D;

    for (int k0 = 0; k0 < D; k0 += KC) {
        __syncthreads();  // previous chunk fully consumed (also publishes linv on iter 0)

        // Stage z chunk: 128 x 64 f32 via float4 (2048 vec4 / 256 threads = 8 each)
        #pragma unroll
        for (int i = 0; i < 8; ++i) {
            int idx = tid + i * 256;
            int b   = idx >> 4;
            int q   = (idx & 15) << 2;
            float4 v = *reinterpret_cast<const float4*>(zt + (size_t)b * D + k0 + q);
            float* dst = &lz[b * LZS + q];
            dst[0] = v.x; dst[1] = v.y; dst[2] = v.z; dst[3] = v.w;
        }
        // Stage normalized A chunk: 64 x 64 f32 (1024 vec4 / 256 threads = 4 each)
        #pragma unroll
        for (int i = 0; i < 4; ++i) {
            int idx = tid + i * 256;
            int d   = idx >> 4;
            int q   = (idx & 15) << 2;
            float4 v = *reinterpret_cast<const float4*>(A + (size_t)(k0 + d) * M + m0 + q);
            float* dst = &la[d * LAS + q];
            dst[0] = v.x * linv[q + 0];
            dst[1] = v.y * linv[q + 1];
            dst[2] = v.z * linv[q + 2];
            dst[3] = v.w * linv[q + 3];
        }
        __syncthreads();

        // WMMA f32 16x16x4 over the chunk. Wave owns rows [16*wave, 16*wave+16).
        // A frag layout: lanes 0-15 = rows, VGPR0/1 = K+0/K+1; lanes 16-31 = K+2/K+3.
        #pragma unroll 4
        for (int kk = 0; kk < KC; kk += 4) {
            const int ka = kk + 2 * half;
            v2f a;
            const float* arow = &lz[(16 * wave + l16) * LZS + ka];
            a.x = arow[0];
            a.y = arow[1];
            #pragma unroll
            for (int ct = 0; ct < 4; ++ct) {
                v2f bb;
                bb.x = la[ka * LAS + ct * 16 + l16];
                bb.y = la[(ka + 1) * LAS + ct * 16 + l16];
                c[ct] = __builtin_amdgcn_wmma_f32_16x16x4_f32(
                    /*neg_a=*/false, a, /*neg_b=*/false, bb,
                    /*c_mod=*/(short)0, c[ct], /*reuse_a=*/false, /*reuse_b=*/false);
            }
        }
    }

    __syncthreads();
    // Spill proj tile to LDS (reuse lz): C layout -> row b = 16*wave + v + 8*half, col = 16*ct + l16
    #pragma unroll
    for (int ct = 0; ct < 4; ++ct) {
        #pragma unroll
        for (int v = 0; v < 8; ++v) {
            int b = 16 * wave + v + 8 * half;
            int m = ct * 16 + l16;
            lz[b * LZS + m] = c[ct][v];
        }
    }
    __syncthreads();

    // Phase 2: each thread owns (m, k) pairs; reduce cos/sin over B with native trans ops.
    const float dt    = 3.0f / 16.0f;
    const float invB  = 1.0f / (float)B;
    float local = 0.f;
    for (int pair = tid; pair < MT * K; pair += 256) {
        int m = pair & (MT - 1);
        int k = pair >> 6;               // MT == 64
        float tk = (float)k * dt;
        float sc = 0.f, ss = 0.f;
        #pragma unroll 4
        for (int b = 0; b < B; ++b) {
            float x = lz[b * LZS + m] * tk;
            sc += __cosf(x);             // v_cos_f32
            ss += __sinf(x);             // v_sin_f32
        }
        float target = __expf(-0.5f * tk * tk);
        float wk = (((k == 0) || (k == K - 1)) ? dt : 2.0f * dt) * target;
        float cm = sc * invB - target;
        float sm = ss * invB;
        local += (cm * cm + sm * sm) * wk;
    }

    red[tid] = local;
    __syncthreads();
    #pragma unroll
    for (int s = 128; s > 0; s >>= 1) {
        if (tid < s) red[tid] += red[tid + s];
        __syncthreads();
    }
    if (tid == 0) {
        // mean over (T,M) of statistic*B  ->  scale = B/(T*M) = 1/256
        atomicAdd(out, red[0] * ((float)B / (float)(T * M)));
    }
}

extern "C" void kernel_launch(void* const* d_in, const int* in_sizes, int n_in,
                              void* d_out, int out_size, void* d_ws, size_t ws_size,
                              hipStream_t stream) {
    const float* z = (const float*)d_in[0];   // (32,128,512) f32
    const float* A = (const float*)d_in[1];   // (512,1024)  f32
    float* out  = (float*)d_out;              // scalar f32
    float* invn = (float*)d_ws;               // 1024 f32 column inverse norms

    col_inv_norm_kernel<<<dim3(M / 256), 256, 0, stream>>>(A, invn);
    zero_kernel<<<1, 1, 0, stream>>>(out);
    sigreg_fused_kernel<<<dim3(T, M / MT), 256, 0, stream>>>(z, A, invn, out);
}